// SphereByCurvature_17274358464573
// MI455X (gfx1250) — compile-verified
//
#include <hip/hip_runtime.h>
#include <hip/hip_bf16.h>

typedef __attribute__((ext_vector_type(2))) float v2f;
typedef __attribute__((ext_vector_type(4))) float f4;
typedef __attribute__((ext_vector_type(8))) float v8f;

#define NUM_ITER 6

__device__ __forceinline__ float sag_sv(float Cc, float y) {
    float r2 = y * y;
    float w  = fmaxf(1.0f - Cc * Cc * r2, 1e-12f);
    float sq = sqrtf(w);
    return Cc * r2 / (1.0f + sq);
}

// ---------------------------------------------------------------------------
// One-wave setup kernel: builds tf_surface / tf_next with V_WMMA_F32_16X16X4_F32
// (4x4 matmuls embedded in the 16x4 x 4x16 shape), writes them to the output
// tail, then computes Minv / b / derived scalars into d_ws for the main pass.
// ---------------------------------------------------------------------------
__global__ __launch_bounds__(32) void setup_tf_kernel(
    const float* __restrict__ tf_in, const float* __restrict__ dia_p,
    const float* __restrict__ C_p,   const float* __restrict__ anch,
    const float* __restrict__ scale_p,
    float* __restrict__ out_tf,   // 32 floats: tf_surface(16) then tf_next(16)
    float* __restrict__ consts)   // ws: Minv(9), b(3), C^2, (d/2)^2, C
{
    const int lane   = threadIdx.x;          // wave32, EXEC all ones at WMMA
    const float dia   = dia_p[0];
    const float Cc    = C_p[0];
    const float scale = scale_p[0];
    const float ext0  = sag_sv(Cc, anch[0] * dia * 0.5f);
    const float ext1  = sag_sv(Cc, anch[1] * dia * 0.5f);
    const float dx1   = -scale * ext0;          // tf_surface translation
    const float dx2   = scale * (ext1 - ext0);  // tf_next translation

    // A fragment: tf_in padded to 16x4 (rows >= 4 zero).
    // ISA layout: lane m=lane&15, VGPR0=K=k0, VGPR1=K=k0+1, k0=(lane>>4)*2.
    const int m  = lane & 15;
    const int k0 = (lane >> 4) * 2;
    v2f a;
    a.x = (m < 4) ? tf_in[m * 4 + k0]     : 0.0f;
    a.y = (m < 4) ? tf_in[m * 4 + k0 + 1] : 0.0f;

    // B fragments (4x16, cols >= 4 zero). lane n=lane&15 indexes columns.
    const int n = lane & 15;
    // R1 = Tx(dx1) @ diag(scale,scale,scale,1): diag with R1[0][3] = dx1
    auto R1 = [&](int i, int j) -> float {
        if (j >= 4) return 0.0f;
        float v = (i == j) ? ((i < 3) ? scale : 1.0f) : 0.0f;
        return (i == 0 && j == 3) ? dx1 : v;
    };
    // R2 = Tx(dx2): identity with R2[0][3] = dx2
    auto R2 = [&](int i, int j) -> float {
        if (j >= 4) return 0.0f;
        float v = (i == j) ? 1.0f : 0.0f;
        return (i == 0 && j == 3) ? dx2 : v;
    };
    v2f bf1; bf1.x = R1(k0, n); bf1.y = R1(k0 + 1, n);
    v2f bf2; bf2.x = R2(k0, n); bf2.y = R2(k0 + 1, n);

    v8f cz = {};
    // D = A x B + 0  (8 args: neg_a, A, neg_b, B, c_mod, C, reuse_a, reuse_b)
    v8f d1 = __builtin_amdgcn_wmma_f32_16x16x4_f32(false, a, false, bf1,
                                                   (short)0, cz, false, false);
    v8f d2 = __builtin_amdgcn_wmma_f32_16x16x4_f32(false, a, false, bf2,
                                                   (short)0, cz, false, false);

    // C/D layout: VGPR v -> row M=v (lanes 0-15), col N = lane.
    __shared__ float sm[16];
    if (lane < 4) {
        #pragma unroll
        for (int v = 0; v < 4; ++v) {
            out_tf[v * 4 + lane]      = d1[v];   // tf_surface
            out_tf[16 + v * 4 + lane] = d2[v];   // tf_next
            sm[v * 4 + lane]          = d1[v];
        }
    }
    __syncthreads();

    if (lane == 0) {
        // M = tf_surface[:3,:3], b = tf_surface[:3,3]; Minv via adjugate.
        float m00 = sm[0], m01 = sm[1], m02 = sm[2];
        float m10 = sm[4], m11 = sm[5], m12 = sm[6];
        float m20 = sm[8], m21 = sm[9], m22 = sm[10];
        float c00 = m11 * m22 - m12 * m21;
        float c01 = m12 * m20 - m10 * m22;
        float c02 = m10 * m21 - m11 * m20;
        float det = m00 * c00 + m01 * c01 + m02 * c02;
        float id  = 1.0f / det;
        consts[0] = c00 * id;                     // i00
        consts[1] = (m02 * m21 - m01 * m22) * id; // i01
        consts[2] = (m01 * m12 - m02 * m11) * id; // i02
        consts[3] = c01 * id;                     // i10
        consts[4] = (m00 * m22 - m02 * m20) * id; // i11
        consts[5] = (m02 * m10 - m00 * m12) * id; // i12
        consts[6] = c02 * id;                     // i20
        consts[7] = (m01 * m20 - m00 * m21) * id; // i21
        consts[8] = (m00 * m11 - m01 * m10) * id; // i22
        consts[9]  = sm[3];
        consts[10] = sm[7];
        consts[11] = sm[11];
        consts[12] = Cc * Cc;
        consts[13] = (dia * 0.5f) * (dia * 0.5f);
        consts[14] = Cc;
    }
}

// ---------------------------------------------------------------------------
// Streaming Newton kernel: 4 rays per thread so all global traffic is b128.
// Memory-bound (44 B/ray, AI ~3.6 flop/B) -> coalesced 128-bit VMEM is the
// optimal MI455X path; no LDS staging (zero reuse).
// ---------------------------------------------------------------------------
__global__ __launch_bounds__(256) void newton_sag_kernel(
    const float* __restrict__ P, const float* __restrict__ Vv,
    const float* __restrict__ consts, float* __restrict__ out,
    int nQuads, long long N)
{
    const int tid = blockIdx.x * blockDim.x + threadIdx.x;
    if (tid >= nQuads) return;

    // Uniform constants -> scalar loads
    const float i00 = consts[0], i01 = consts[1], i02 = consts[2];
    const float i10 = consts[3], i11 = consts[4], i12 = consts[5];
    const float i20 = consts[6], i21 = consts[7], i22 = consts[8];
    const float b0  = consts[9], b1  = consts[10], b2 = consts[11];
    const float C2  = consts[12], halfD2 = consts[13], Cc = consts[14];

    const f4* __restrict__ P4 = (const f4*)P;
    const f4* __restrict__ V4 = (const f4*)Vv;

    // Speculative prefetch ~2 MB ahead (global_prefetch_b8; dropped if unmapped)
    {
        const long long pf = (long long)tid * 3 + 3LL * 43690;
        __builtin_prefetch(&P4[pf], 0, 1);
        __builtin_prefetch(&V4[pf], 0, 1);
    }

    const long long base = (long long)tid * 3;
    f4 pA = P4[base + 0], pB = P4[base + 1], pC = P4[base + 2];
    f4 vA = V4[base + 0], vB = V4[base + 1], vC = V4[base + 2];

    float pr[12] = {pA.x, pA.y, pA.z, pA.w, pB.x, pB.y, pB.z, pB.w,
                    pC.x, pC.y, pC.z, pC.w};
    float vr[12] = {vA.x, vA.y, vA.z, vA.w, vB.x, vB.y, vB.z, vB.w,
                    vC.x, vC.y, vC.z, vC.w};

    float tOut[4], valOut[4], nOut[12];

    #pragma unroll
    for (int r = 0; r < 4; ++r) {
        const float px = pr[3 * r + 0] - b0;
        const float py = pr[3 * r + 1] - b1;
        const float pz = pr[3 * r + 2] - b2;
        const float vx = vr[3 * r + 0];
        const float vy = vr[3 * r + 1];
        const float vz = vr[3 * r + 2];

        // Pl = Minv @ (P - b); Vl = Minv @ V
        const float pl0 = i00 * px + i01 * py + i02 * pz;
        const float pl1 = i10 * px + i11 * py + i12 * pz;
        const float pl2 = i20 * px + i21 * py + i22 * pz;
        const float vl0 = i00 * vx + i01 * vy + i02 * vz;
        const float vl1 = i10 * vx + i11 * vy + i12 * vz;
        const float vl2 = i20 * vx + i21 * vy + i22 * vz;

        float t = 0.0f;
        #pragma unroll
        for (int it = 0; it < NUM_ITER; ++it) {
            const float q0 = fmaf(t, vl0, pl0);
            const float q1 = fmaf(t, vl1, pl1);
            const float q2 = fmaf(t, vl2, pl2);
            const float r2 = q1 * q1 + q2 * q2;
            const float w  = fmaxf(1.0f - C2 * r2, 1e-12f);
            const float sq = sqrtf(w);
            const float dd = Cc / sq;
            const float sv = Cc * r2 / (1.0f + sq);
            const float F  = sv - q0;
            float denom = -vl0 + dd * (q1 * vl1 + q2 * vl2);
            denom = (fabsf(denom) > 1e-12f) ? denom : 1e-12f;
            t = t - F / denom;
        }

        // Final evaluation at converged t
        const float q0 = fmaf(t, vl0, pl0);
        const float q1 = fmaf(t, vl1, pl1);
        const float q2 = fmaf(t, vl2, pl2);
        const float r2 = q1 * q1 + q2 * q2;
        const float w  = fmaxf(1.0f - C2 * r2, 1e-12f);
        const float sq = sqrtf(w);
        const float dd = Cc / sq;
        const float sv = Cc * r2 / (1.0f + sq);
        const float F  = sv - q0;
        const float g0 = -1.0f, g1 = dd * q1, g2 = dd * q2;
        const bool ok = (r2 <= halfD2) && (fabsf(F) < 0.001f) && (C2 * r2 < 1.0f);

        const float gn  = 1.0f / sqrtf(g0 * g0 + g1 * g1 + g2 * g2);
        const float nl0 = g0 * gn, nl1 = g1 * gn, nl2 = g2 * gn;
        // n = nl @ Minv  => n_j = sum_i nl_i * Minv[i][j]
        float n0 = nl0 * i00 + nl1 * i10 + nl2 * i20;
        float n1 = nl0 * i01 + nl1 * i11 + nl2 * i21;
        float n2 = nl0 * i02 + nl1 * i12 + nl2 * i22;
        const float nn = 1.0f / sqrtf(n0 * n0 + n1 * n1 + n2 * n2);
        n0 *= nn; n1 *= nn; n2 *= nn;

        tOut[r]   = t;
        valOut[r] = ok ? 1.0f : 0.0f;
        nOut[3 * r + 0] = n0;
        nOut[3 * r + 1] = n1;
        nOut[3 * r + 2] = n2;
    }

    f4* __restrict__ outT = (f4*)out;                 // t:     [0, N)
    f4* __restrict__ outN = (f4*)(out + N);           // n:     [N, 4N)
    f4* __restrict__ outV = (f4*)(out + 4 * N);       // valid: [4N, 5N)

    f4 tv = {tOut[0], tOut[1], tOut[2], tOut[3]};
    f4 n0v = {nOut[0], nOut[1], nOut[2], nOut[3]};
    f4 n1v = {nOut[4], nOut[5], nOut[6], nOut[7]};
    f4 n2v = {nOut[8], nOut[9], nOut[10], nOut[11]};
    f4 vv = {valOut[0], valOut[1], valOut[2], valOut[3]};

    outT[tid]          = tv;
    outN[base + 0]     = n0v;
    outN[base + 1]     = n1v;
    outN[base + 2]     = n2v;
    outV[tid]          = vv;
}

// ---------------------------------------------------------------------------
extern "C" void kernel_launch(void* const* d_in, const int* in_sizes, int n_in,
                              void* d_out, int out_size, void* d_ws, size_t ws_size,
                              hipStream_t stream) {
    const float* P     = (const float*)d_in[0];
    const float* V     = (const float*)d_in[1];
    const float* tf_in = (const float*)d_in[2];
    const float* dia   = (const float*)d_in[3];
    const float* Cp    = (const float*)d_in[4];
    const float* anch  = (const float*)d_in[5];
    const float* scale = (const float*)d_in[6];

    const long long N = (long long)in_sizes[0] / 3;   // P is (N,3)
    float* out    = (float*)d_out;
    float* consts = (float*)d_ws;

    // Stage 1: transform chain via WMMA + Minv/b into workspace.
    setup_tf_kernel<<<1, 32, 0, stream>>>(tf_in, dia, Cp, anch, scale,
                                          out + 5 * N, consts);

    // Stage 2: streaming Newton solve, 4 rays/thread (N is a multiple of 4).
    const int nQuads = (int)(N / 4);
    const int block  = 256;
    const int grid   = (nQuads + block - 1) / block;
    newton_sag_kernel<<<grid, block, 0, stream>>>(P, V, consts, out, nQuads, N);
}